// HYM_85194971284097
// MI455X (gfx1250) — compile-verified
//
#include <hip/hip_runtime.h>

// Sizes from the reference
#define N_ROWS 64
#define C_DIM  128
#define K_DIM  32768
#define J_MAX  10          // Taylor degree for exp(x), |x|<1 -> err <= e/11! ~ 7e-8

static constexpr float INV_T = 1.0f / 0.07f;
static constexpr float EPS_N = 1e-12f;

// 1/j! coefficients (constexpr -> folded to inline immediates after unroll)
static constexpr float INVFACT[J_MAX + 1] = {
    1.0f, 1.0f, 0.5f, 1.0f / 6.0f, 1.0f / 24.0f, 1.0f / 120.0f,
    1.0f / 720.0f, 1.0f / 5040.0f, 1.0f / 40320.0f,
    1.0f / 362880.0f, 1.0f / 3628800.0f};

typedef float v2f __attribute__((ext_vector_type(2)));
typedef float v8f __attribute__((ext_vector_type(8)));

// ---------------------------------------------------------------------------
// l_pos: prob = normalize(ce,dim=1); l_pos[n] = logsumexp_c(dist*prob)/T
// Tiny: 64 rows x 128 cols. One thread per row.
// ---------------------------------------------------------------------------
__global__ __launch_bounds__(64) void lpos_kernel(
    const float* __restrict__ ce, const float* __restrict__ dist,
    float* __restrict__ out) {
  const int n = threadIdx.x;
  const float* cr = ce + n * C_DIM;
  const float* dr = dist + n * C_DIM;

  float ss = 0.0f;
  for (int c = 0; c < C_DIM; ++c) { float v = cr[c]; ss += v * v; }
  const float inv = 1.0f / fmaxf(sqrtf(ss), EPS_N);

  float mx = -INFINITY;
  for (int c = 0; c < C_DIM; ++c) mx = fmaxf(mx, dr[c] * cr[c] * inv);
  float se = 0.0f;
  for (int c = 0; c < C_DIM; ++c) se += __expf(dr[c] * cr[c] * inv - mx);

  out[(size_t)n * (K_DIM + 1)] = (__logf(se) + mx) * INV_T;
}

// ---------------------------------------------------------------------------
// l_neg via Taylor-factorized logsumexp:
//   S[n,k] = C + sum_{j=1..J} (1/j!) * (dist^j @ bnorm^j)[n,k]
//   out    = log(S)/T
// Block = 128 threads (4 waves), owns a 16-wide k strip. Wave w computes the
// 16x16 tile with rows [16w,16w+16). All GEMM work on v_wmma_f32_16x16x4_f32.
// ---------------------------------------------------------------------------
__global__ __launch_bounds__(128) void lneg_kernel(
    const float* __restrict__ dist, const float* __restrict__ queue,
    float* __restrict__ out) {
  __shared__ float ldsD[N_ROWS * 129];   // dist, row stride 129 (bank-pad)
  __shared__ float ldsB[C_DIM * 16];     // normalized queue strip [c][k]
  __shared__ float ldsInv[16];

  const int t  = threadIdx.x;
  const int k0 = blockIdx.x << 4;

  // Stage dist [64][128] -> LDS (padded). Hits L2 after first block.
  for (int idx = t; idx < N_ROWS * C_DIM; idx += 128) {
    const int r = idx >> 7, c = idx & 127;
    ldsD[r * 129 + c] = dist[idx];
  }

  // Stage queue strip: thread t loads row c=t, 16 contiguous floats.
  {
    const float* src = queue + (size_t)t * K_DIM + k0;
    float4* dst = (float4*)&ldsB[t * 16];
    const float4* s4 = (const float4*)src;
    dst[0] = s4[0]; dst[1] = s4[1]; dst[2] = s4[2]; dst[3] = s4[3];
  }
  __syncthreads();

  // Column L2 norms over C (axis 0), 16 columns.
  if (t < 16) {
    float s = 0.0f;
    for (int c = 0; c < C_DIM; ++c) { float v = ldsB[c * 16 + t]; s += v * v; }
    ldsInv[t] = 1.0f / fmaxf(sqrtf(s), EPS_N);
  }
  __syncthreads();

  // Normalize strip in place.
  for (int idx = t; idx < C_DIM * 16; idx += 128)
    ldsB[idx] *= ldsInv[idx & 15];
  __syncthreads();

  // --- WMMA tile compute ---
  const int wave = t >> 5;
  const int lane = t & 31;
  const int hi   = lane >> 4;              // half-wave
  const int off  = hi << 1;                // A/B K-slot: 0 or 2
  const int mA   = (wave << 4) + (lane & 15);  // A-fragment row
  const int n    = lane & 15;              // B/D column within tile

  v8f acc;
#pragma unroll
  for (int e = 0; e < 8; ++e) acc[e] = (float)C_DIM;   // j = 0 term: sum_c 1

  const float* dRow = &ldsD[mA * 129];

  for (int c0 = 0; c0 < C_DIM; c0 += 4) {
    // A 16x4 f32 layout: lanes 0-15 hold K=0(v0),1(v1); lanes 16-31 K=2,3.
    const float a0 = dRow[c0 + off];
    const float a1 = dRow[c0 + off + 1];
    // B 4x16 f32 (mirrored striping): row K, col = lane%16.
    const float b0 = ldsB[(c0 + off) * 16 + n];
    const float b1 = ldsB[(c0 + off + 1) * 16 + n];

    float ap0 = a0, ap1 = a1, bp0 = b0, bp1 = b1;
    v2f A, B;
    A[0] = ap0; A[1] = ap1; B[0] = bp0; B[1] = bp1;
    // j = 1 (coef 1)
    acc = __builtin_amdgcn_wmma_f32_16x16x4_f32(
        false, A, false, B, (short)0, acc, false, false);
#pragma unroll
    for (int j = 2; j <= J_MAX; ++j) {
      ap0 *= a0; ap1 *= a1; bp0 *= b0; bp1 *= b1;
      const float cf = INVFACT[j];
      A[0] = ap0 * cf; A[1] = ap1 * cf; B[0] = bp0; B[1] = bp1;
      acc = __builtin_amdgcn_wmma_f32_16x16x4_f32(
          false, A, false, B, (short)0, acc, false, false);
    }
  }

  // D 16x16 f32 layout: VGPR e -> row e (lanes 0-15) / row e+8 (lanes 16-31).
  const int rowBase = (wave << 4) + (hi << 3);
  const int col = k0 + n;
#pragma unroll
  for (int e = 0; e < 8; ++e) {
    const size_t row = (size_t)(rowBase + e);
    out[row * (K_DIM + 1) + 1 + col] = __logf(acc[e]) * INV_T;
  }
}

// ---------------------------------------------------------------------------
extern "C" void kernel_launch(void* const* d_in, const int* in_sizes, int n_in,
                              void* d_out, int out_size, void* d_ws,
                              size_t ws_size, hipStream_t stream) {
  (void)in_sizes; (void)n_in; (void)d_ws; (void)ws_size; (void)out_size;
  const float* ce    = (const float*)d_in[0];   // [64,128]
  const float* dist  = (const float*)d_in[1];   // [64,128]
  const float* queue = (const float*)d_in[2];   // [128,32768]
  float* out = (float*)d_out;                   // [64,32769]

  lpos_kernel<<<1, 64, 0, stream>>>(ce, dist, out);
  lneg_kernel<<<K_DIM / 16, 128, 0, stream>>>(dist, queue, out);
}